// GCMCGraphConv_88029649699239
// MI455X (gfx1250) — compile-verified
//
#include <hip/hip_runtime.h>

#define OUT_DIM 64

// ---------------------------------------------------------------------------
// Kernel A (placed FIRST in the module so the disasm snippet shows it):
// per-edge gather + scatter-add.
//   msg[e]  = weight[node_ids[src_idx[e]]] * cj[src_idx[e]]
//   out[d] += sum_e msg[e]          (segment sum over dst_idx)
//   final   = out * ci              -> folded as per-edge scale cj[s]*ci[d]
//
// Mapping: 16 lanes per edge, lane k owns columns [4k, 4k+4).
//  - weight row read as one aligned float4 (global_load_b128), 256 B per edge
//    fully coalesced across the 16-lane group (wave32 covers 2 edges).
//  - 4 non-returning relaxed agent-scope f32 atomic adds per lane
//    (global_atomic_add_f32, contiguous addresses within the group).
//  - index / cj / ci loads are group-uniform -> broadcast by the coalescer.
// Working set (feat 25.6MB + out 25.6MB + indices 25.6MB) is L2-resident
// on the 192MB L2; the ceiling is L2 f32-atomic throughput, not HBM.
// ---------------------------------------------------------------------------
__global__ __launch_bounds__(256) void GCMC_edge_scatter_kernel(
    const int*   __restrict__ node_ids,
    const int*   __restrict__ src_idx,
    const int*   __restrict__ dst_idx,
    const float* __restrict__ cj,
    const float* __restrict__ ci,
    const float* __restrict__ weight,
    float*                    out,
    unsigned int              n_edges) {
    unsigned int gid  = blockIdx.x * blockDim.x + threadIdx.x;
    unsigned int e    = gid >> 4;        // edge id
    if (e >= n_edges) return;
    unsigned int lane = gid & 15u;       // 16 lanes per edge

    int s   = src_idx[e];
    int d   = dst_idx[e];
    int row = node_ids[s];               // identity in practice, honored anyway

    float scale = cj[s] * ci[d];         // sum(m)*ci == sum(m*ci), ci const per d

    const float4* wrow = reinterpret_cast<const float4*>(
        weight + (size_t)row * OUT_DIM);
    float4 w = wrow[lane];               // global_load_b128

    float* o = out + (size_t)d * OUT_DIM + lane * 4u;

    // Non-returning f32 atomic adds -> global_atomic_add_f32 (no RET),
    // tracked with STOREcnt, resolved at L2.
    __hip_atomic_fetch_add(o + 0, w.x * scale, __ATOMIC_RELAXED, __HIP_MEMORY_SCOPE_AGENT);
    __hip_atomic_fetch_add(o + 1, w.y * scale, __ATOMIC_RELAXED, __HIP_MEMORY_SCOPE_AGENT);
    __hip_atomic_fetch_add(o + 2, w.z * scale, __ATOMIC_RELAXED, __HIP_MEMORY_SCOPE_AGENT);
    __hip_atomic_fetch_add(o + 3, w.w * scale, __ATOMIC_RELAXED, __HIP_MEMORY_SCOPE_AGENT);
}

// ---------------------------------------------------------------------------
// Kernel B: zero the output accumulator (must run every call; the atomics
// above accumulate into it and the harness does not re-zero between replays).
// Vectorized 128-bit stores (global_store_b128).
// ---------------------------------------------------------------------------
__global__ __launch_bounds__(256) void GCMC_zero_out_kernel(float4* __restrict__ out,
                                                            unsigned int n4) {
    unsigned int i = blockIdx.x * blockDim.x + threadIdx.x;
    if (i < n4) {
        out[i] = make_float4(0.f, 0.f, 0.f, 0.f);
    }
}

// ---------------------------------------------------------------------------
// Host-side launch.
// Input order (from setup_inputs): node_ids, src_idx, dst_idx, cj, ci, weight.
// ---------------------------------------------------------------------------
extern "C" void kernel_launch(void* const* d_in, const int* in_sizes, int n_in,
                              void* d_out, int out_size, void* d_ws, size_t ws_size,
                              hipStream_t stream) {
    const int*   node_ids = (const int*)  d_in[0];
    const int*   src_idx  = (const int*)  d_in[1];
    const int*   dst_idx  = (const int*)  d_in[2];
    const float* cj       = (const float*)d_in[3];
    const float* ci       = (const float*)d_in[4];
    const float* weight   = (const float*)d_in[5];
    float*       out      = (float*)      d_out;

    const unsigned int n_edges = (unsigned int)in_sizes[1];
    const unsigned int n_dst   = (unsigned int)in_sizes[4];

    // 1) zero accumulator: n_dst * 64 floats = n_dst * 16 float4s
    {
        unsigned int n4     = n_dst * (OUT_DIM / 4);
        unsigned int blocks = (n4 + 255u) / 256u;
        GCMC_zero_out_kernel<<<blocks, 256, 0, stream>>>((float4*)out, n4);
    }

    // 2) gather + scatter-add: 16 threads per edge
    {
        unsigned long long threads = (unsigned long long)n_edges * 16ull;
        unsigned int blocks = (unsigned int)((threads + 255ull) / 256ull);
        GCMC_edge_scatter_kernel<<<blocks, 256, 0, stream>>>(
            node_ids, src_idx, dst_idx, cj, ci, weight, out, n_edges);
    }
}